// MyVoxelSetAbstraction_53102975648079
// MI455X (gfx1250) — compile-verified
//
#include <hip/hip_runtime.h>

// ---------------------------------------------------------------- constants
#define BB      2
#define N_RAW   16384
#define N_KP    2048
#define M_VOX   8192
#define H_BEV   200
#define W_BEV   176
#define C_BEV   256
#define F_DIM   416     // 256 bev | 32 raw | 128 conv
#define OUT_DIM 128
#define BN_SCALE 0.9999950000374997f

typedef __attribute__((ext_vector_type(2))) float v2f;
typedef __attribute__((ext_vector_type(8))) float v8f;

static __device__ __forceinline__ v8f wmma_k4(v2f a, v2f b, v8f c) {
  // D = A(16x4 f32) * B(4x16 f32) + C(16x16 f32)  -- full fp32 matrix op on CDNA5
  return __builtin_amdgcn_wmma_f32_16x16x4_f32(false, a, false, b, (short)0, c,
                                               false, false);
}
static __device__ __forceinline__ void lds_fence() {
  asm volatile("s_wait_dscnt 0" ::: "memory");
}
static __device__ __forceinline__ v8f zero8() {
  v8f z = {0.f, 0.f, 0.f, 0.f, 0.f, 0.f, 0.f, 0.f};
  return z;
}

// ---------------------------------------------------------------- 1) FPS
// One 1024-thread block per batch; each thread owns 16 contiguous points
// (coords + running min-distance in registers). Sequential over 2047 picks;
// argmax via wave32 shuffles + LDS; ties -> smallest index (jnp.argmax rule).
__global__ __launch_bounds__(1024) void fps_kernel(const float* __restrict__ points,
                                                   float* __restrict__ kp) {
  const int b = blockIdx.x;
  const int t = threadIdx.x;
  const float* P = points + (size_t)b * N_RAW * 5;

  float px[16], py[16], pz[16], dd[16];
#pragma unroll
  for (int j = 0; j < 16; ++j) {
    int i = t * 16 + j;
    px[j] = P[i * 5 + 1];
    py[j] = P[i * 5 + 2];
    pz[j] = P[i * 5 + 3];
    dd[j] = 1e10f;
  }

  __shared__ float s_best[32];
  __shared__ int s_bidx[32];
  __shared__ int s_sel;

  if (t == 0) {
    float* o = kp + (size_t)b * N_KP * 3;
    o[0] = P[1]; o[1] = P[2]; o[2] = P[3];
  }

  int last = 0;
  for (int k = 1; k < N_KP; ++k) {
    const float lx = P[last * 5 + 1];
    const float ly = P[last * 5 + 2];
    const float lz = P[last * 5 + 3];
    float best = -1.0f;
    int bidx = 0x7fffffff;
#pragma unroll
    for (int j = 0; j < 16; ++j) {
      float dx = px[j] - lx, dy = py[j] - ly, dz = pz[j] - lz;
      float d = dx * dx + dy * dy + dz * dz;
      d = fminf(dd[j], d);
      dd[j] = d;
      if (d > best) { best = d; bidx = t * 16 + j; }
    }
#pragma unroll
    for (int off = 16; off > 0; off >>= 1) {
      float ob = __shfl_xor(best, off, 32);
      int oi = __shfl_xor(bidx, off, 32);
      if (ob > best || (ob == best && oi < bidx)) { best = ob; bidx = oi; }
    }
    if ((t & 31) == 0) { s_best[t >> 5] = best; s_bidx[t >> 5] = bidx; }
    __syncthreads();
    if (t < 32) {
      best = s_best[t];
      bidx = s_bidx[t];
#pragma unroll
      for (int off = 16; off > 0; off >>= 1) {
        float ob = __shfl_xor(best, off, 32);
        int oi = __shfl_xor(bidx, off, 32);
        if (ob > best || (ob == best && oi < bidx)) { best = ob; bidx = oi; }
      }
      if (t == 0) {
        s_sel = bidx;
        float* o = kp + ((size_t)b * N_KP + k) * 3;
        o[0] = P[bidx * 5 + 1];
        o[1] = P[bidx * 5 + 2];
        o[2] = P[bidx * 5 + 3];
      }
    }
    __syncthreads();
    last = s_sel;
  }
}

// ---------------------------------------------------------------- 2) BEV bilinear
// block = 256 threads = one keypoint; thread = one channel.
__global__ __launch_bounds__(256) void bev_kernel(const float* __restrict__ sf,
                                                  const float* __restrict__ kp,
                                                  float* __restrict__ feats) {
  const int r = blockIdx.x;          // 0 .. B*N_KP-1
  const int b = r / N_KP;
  const int c = threadIdx.x;
  const float x = kp[r * 3 + 0];
  const float y = kp[r * 3 + 1];
  const float xi = (x - 0.0f) / 0.05f / 8.0f;
  const float yi = (y - (-40.0f)) / 0.05f / 8.0f;
  const float xf = floorf(xi), yf = floorf(yi);
  int x0 = (int)xf, y0 = (int)yf;
  int x1 = x0 + 1, y1 = y0 + 1;
  x0 = min(max(x0, 0), W_BEV - 1);
  x1 = min(max(x1, 0), W_BEV - 1);
  y0 = min(max(y0, 0), H_BEV - 1);
  y1 = min(max(y1, 0), H_BEV - 1);
  const float* base = sf + ((size_t)b * C_BEV + c) * (H_BEV * W_BEV);
  const float fd = base[y0 * W_BEV + x0];
  const float fc_ = base[y0 * W_BEV + x1];
  const float fa = base[y1 * W_BEV + x0];
  const float fb = base[y1 * W_BEV + x1];
  const float x0f = (float)x0, x1f = (float)x1, y0f = (float)y0, y1f = (float)y1;
  const float wa = (xi - x0f) * (y1f - yi);
  const float wb = (x1f - xi) * (y1f - yi);
  const float wc = (x1f - xi) * (yi - y0f);
  const float wd = (xi - x0f) * (yi - y0f);
  feats[(size_t)r * F_DIM + c] = fd * wb + fc_ * wa + fa * wc + fb * wd;
}

// ---------------------------------------------------------------- 3) ball query (raw points)
// one wave per (keypoint, radius); scan in index order, keep first ns hits.
__global__ __launch_bounds__(256) void raw_group_kernel(const float* __restrict__ points,
                                                        const float* __restrict__ kp,
                                                        int* __restrict__ ridx,
                                                        int* __restrict__ rcnt) {
  const int wid = threadIdx.x >> 5;
  const int lane = threadIdx.x & 31;
  const int g = blockIdx.x * 8 + wid;
  const int kpg = g >> 1;
  const int ri = g & 1;
  const int b = kpg / N_KP;
  const float radius = ri ? 0.8f : 0.4f;
  const float r2 = radius * radius;
  const int ns = 16;
  const float* P = points + (size_t)b * N_RAW * 5;
  const float cx = kp[kpg * 3 + 0], cy = kp[kpg * 3 + 1], cz = kp[kpg * 3 + 2];
  int* my = ridx + (size_t)kpg * 32 + ri * 16;
  int cnt = 0, first = -1;
  for (int base = 0; base < N_RAW && cnt < ns; base += 32) {
    const int i = base + lane;
    const float dx = P[i * 5 + 1] - cx;
    const float dy = P[i * 5 + 2] - cy;
    const float dz = P[i * 5 + 3] - cz;
    const bool in = (dx * dx + dy * dy + dz * dz) < r2;
    const unsigned mask = (unsigned)__ballot(in ? 1 : 0);
    if (first < 0 && mask) first = base + (__ffs((int)mask) - 1);
    const int rank = __popc(mask & ((1u << lane) - 1u));
    if (in && (cnt + rank) < ns) my[cnt + rank] = i;
    cnt += __popc(mask);
  }
  if (cnt > ns) cnt = ns;
  const int pad = (first < 0) ? 0 : first;
  for (int p = cnt + lane; p < ns; p += 32) my[p] = pad;
  if (lane == 0) rcnt[kpg * 2 + ri] = cnt;
}

// ---------------------------------------------------------------- 4) ball query (voxel centers)
__global__ __launch_bounds__(256) void conv_group_kernel(const int* __restrict__ vcoords,
                                                         const float* __restrict__ kp,
                                                         int* __restrict__ cidx,
                                                         int* __restrict__ ccnt) {
  const int wid = threadIdx.x >> 5;
  const int lane = threadIdx.x & 31;
  const int g = blockIdx.x * 8 + wid;
  const int kpg = g >> 1;
  const int ri = g & 1;
  const int b = kpg / N_KP;
  const float radius = ri ? 2.4f : 1.2f;
  const float r2 = radius * radius;
  const int ns = ri ? 32 : 16;
  const int* VC = vcoords + (size_t)b * M_VOX * 4;
  const float cx = kp[kpg * 3 + 0], cy = kp[kpg * 3 + 1], cz = kp[kpg * 3 + 2];
  int* my = cidx + (size_t)kpg * 48 + ri * 16;
  int cnt = 0, first = -1;
  for (int base = 0; base < M_VOX && cnt < ns; base += 32) {
    const int i = base + lane;
    const int ix = VC[i * 4 + 3], iy = VC[i * 4 + 2], iz = VC[i * 4 + 1];
    const float X = ((float)ix + 0.5f) * 0.2f + 0.0f;
    const float Y = ((float)iy + 0.5f) * 0.2f + (-40.0f);
    const float Z = ((float)iz + 0.5f) * 0.4f + (-3.0f);
    const float dx = X - cx, dy = Y - cy, dz = Z - cz;
    const bool in = (dx * dx + dy * dy + dz * dz) < r2;
    const unsigned mask = (unsigned)__ballot(in ? 1 : 0);
    if (first < 0 && mask) first = base + (__ffs((int)mask) - 1);
    const int rank = __popc(mask & ((1u << lane) - 1u));
    if (in && (cnt + rank) < ns) my[cnt + rank] = i;
    cnt += __popc(mask);
  }
  if (cnt > ns) cnt = ns;
  const int pad = (first < 0) ? 0 : first;
  for (int p = cnt + lane; p < ns; p += 32) my[p] = pad;
  if (lane == 0) ccnt[kpg * 2 + ri] = cnt;
}

// ---------------------------------------------------------------- 5) raw SA MLP (WMMA fp32)
// one wave per (keypoint, radius): M=16 samples, 4->16->16, max-pool over M.
__global__ __launch_bounds__(128) void raw_mlp_kernel(
    const float* __restrict__ points, const float* __restrict__ kp,
    const int* __restrict__ ridx, const int* __restrict__ rcnt,
    const float* __restrict__ w0, const float* __restrict__ g0, const float* __restrict__ b0,
    const float* __restrict__ w1, const float* __restrict__ g1, const float* __restrict__ b1,
    float* __restrict__ feats) {
  __shared__ __align__(16) float hbuf[4][16 * 16];
  const int wid = threadIdx.x >> 5;
  const int lane = threadIdx.x & 31;
  const int g = blockIdx.x * 4 + wid;
  const int kpg = g >> 1;
  const int ri = g & 1;
  const int b = kpg / N_KP;
  const int m = lane & 15;            // A: row (sample); B/C: column
  const int koff = (lane < 16) ? 0 : 2;
  const float* P = points + (size_t)b * N_RAW * 5;
  const float cx = kp[kpg * 3], cy = kp[kpg * 3 + 1], cz = kp[kpg * 3 + 2];

  // gather one sample per lane pair into A layout (k = koff, koff+1)
  const int sidx = ridx[(size_t)kpg * 32 + ri * 16 + m];
  const float* pr = P + (size_t)sidx * 5;
  const float dx = pr[1] - cx, dy = pr[2] - cy, dz = pr[3] - cz, it = pr[4];
  v2f a;
  a.x = (lane < 16) ? dx : dz;
  a.y = (lane < 16) ? dy : it;

  const float* W0 = w0 + ri * 4 * 16;   // (4,16) row-major
  v2f bm;
  bm.x = W0[(koff + 0) * 16 + m];
  bm.y = W0[(koff + 1) * 16 + m];
  v8f c = zero8();
  c = wmma_k4(a, bm, c);

  float gs = g0[ri * 16 + m] * BN_SCALE;
  float bb = b0[ri * 16 + m];
#pragma unroll
  for (int v = 0; v < 8; ++v) c[v] = fmaxf(c[v] * gs + bb, 0.f);

  // re-stage h (C layout -> A layout) through wave-private LDS
  float* hb = hbuf[wid];
  const int mbase = (lane < 16) ? 0 : 8;
#pragma unroll
  for (int v = 0; v < 8; ++v) hb[(mbase + v) * 16 + m] = c[v];
  lds_fence();

  const float* W1 = w1 + ri * 16 * 16;  // (16,16)
  v8f c2 = zero8();
#pragma unroll
  for (int k0 = 0; k0 < 16; k0 += 4) {
    const float2 av = *(const float2*)(hb + m * 16 + k0 + koff);
    v2f a2; a2.x = av.x; a2.y = av.y;
    v2f b2;
    b2.x = W1[(k0 + koff + 0) * 16 + m];
    b2.y = W1[(k0 + koff + 1) * 16 + m];
    c2 = wmma_k4(a2, b2, c2);
  }
  gs = g1[ri * 16 + m] * BN_SCALE;
  bb = b1[ri * 16 + m];
  float mx = -1e30f;
#pragma unroll
  for (int v = 0; v < 8; ++v) mx = fmaxf(mx, fmaxf(c2[v] * gs + bb, 0.f));
  mx = fmaxf(mx, __shfl_xor(mx, 16, 32));
  if (rcnt[kpg * 2 + ri] == 0) mx = 0.f;
  if (lane < 16) feats[(size_t)kpg * F_DIM + 256 + ri * 16 + m] = mx;
}

// ---------------------------------------------------------------- 6) conv SA MLP (WMMA fp32)
// one wave per (keypoint, radius): M=ns (16/32), 67(+pad)->64->64, pool over M.
__global__ __launch_bounds__(64) void conv_mlp_kernel(
    const int* __restrict__ vcoords, const float* __restrict__ vfeats,
    const float* __restrict__ kp,
    const int* __restrict__ cidx, const int* __restrict__ ccnt,
    const float* __restrict__ w0, const float* __restrict__ g0, const float* __restrict__ b0,
    const float* __restrict__ w1, const float* __restrict__ g1, const float* __restrict__ b1,
    float* __restrict__ feats) {
  __shared__ __align__(16) float inbuf[2][32 * 68];
  __shared__ __align__(16) float hbuf[2][32 * 64];
  const int wid = threadIdx.x >> 5;
  const int lane = threadIdx.x & 31;
  const int g = blockIdx.x * 2 + wid;
  const int kpg = g >> 1;
  const int ri = g & 1;
  const int b = kpg / N_KP;
  const int ns = ri ? 32 : 16;
  const int mt_n = ns >> 4;
  const float cx = kp[kpg * 3], cy = kp[kpg * 3 + 1], cz = kp[kpg * 3 + 2];
  const int* VC = vcoords + (size_t)b * M_VOX * 4;
  const float* VF = vfeats + (size_t)b * M_VOX * 64;
  float* ibuf = inbuf[wid];

  // stage grouped features (3 rel-coords + 64 voxel feats + 1 zero pad) into LDS
  for (int s = lane; s < ns; s += 32) {
    const int idx = cidx[(size_t)kpg * 48 + ri * 16 + s];
    const int ix = VC[idx * 4 + 3], iy = VC[idx * 4 + 2], iz = VC[idx * 4 + 1];
    const float X = ((float)ix + 0.5f) * 0.2f + 0.0f;
    const float Y = ((float)iy + 0.5f) * 0.2f + (-40.0f);
    const float Z = ((float)iz + 0.5f) * 0.4f + (-3.0f);
    float* row = ibuf + s * 68;
    row[0] = X - cx; row[1] = Y - cy; row[2] = Z - cz;
    const float* vf = VF + (size_t)idx * 64;
    for (int f = 0; f < 64; ++f) row[3 + f] = vf[f];
    row[67] = 0.f;
  }
  lds_fence();

  const int m = lane & 15;
  const int koff = (lane < 16) ? 0 : 2;
  const float* W0 = w0 + (size_t)ri * 67 * 64;  // (67,64)
  float* hb = hbuf[wid];

  // layer 1: K = 68 (row 67 is zero pad in A; clamp B row index to stay in-bounds)
  for (int mt = 0; mt < mt_n; ++mt) {
    const int row = mt * 16 + m;
    for (int nt = 0; nt < 4; ++nt) {
      v8f acc = zero8();
      for (int k0 = 0; k0 < 68; k0 += 4) {
        const float2 av = *(const float2*)(ibuf + row * 68 + k0 + koff);
        v2f a; a.x = av.x; a.y = av.y;
        const int ka = k0 + koff;
        const int kac = (ka > 66) ? 66 : ka;
        const int kbc = (ka + 1 > 66) ? 66 : (ka + 1);
        v2f bm;
        bm.x = W0[(size_t)kac * 64 + nt * 16 + m];
        bm.y = W0[(size_t)kbc * 64 + nt * 16 + m];
        acc = wmma_k4(a, bm, acc);
      }
      const float gs = g0[ri * 64 + nt * 16 + m] * BN_SCALE;
      const float bv = b0[ri * 64 + nt * 16 + m];
      const int mbase = mt * 16 + ((lane < 16) ? 0 : 8);
#pragma unroll
      for (int v = 0; v < 8; ++v)
        hb[(mbase + v) * 64 + nt * 16 + m] = fmaxf(acc[v] * gs + bv, 0.f);
    }
  }
  lds_fence();

  // layer 2: K = 64, then column-max pool over all ns rows
  const float* W1 = w1 + (size_t)ri * 64 * 64;  // (64,64)
  const int cnt = ccnt[kpg * 2 + ri];
  for (int nt = 0; nt < 4; ++nt) {
    const float gs = g1[ri * 64 + nt * 16 + m] * BN_SCALE;
    const float bv = b1[ri * 64 + nt * 16 + m];
    float colmax = -1e30f;
    for (int mt = 0; mt < mt_n; ++mt) {
      const int row = mt * 16 + m;
      v8f acc = zero8();
      for (int k0 = 0; k0 < 64; k0 += 4) {
        const float2 av = *(const float2*)(hb + row * 64 + k0 + koff);
        v2f a; a.x = av.x; a.y = av.y;
        v2f bm;
        bm.x = W1[(size_t)(k0 + koff + 0) * 64 + nt * 16 + m];
        bm.y = W1[(size_t)(k0 + koff + 1) * 64 + nt * 16 + m];
        acc = wmma_k4(a, bm, acc);
      }
#pragma unroll
      for (int v = 0; v < 8; ++v)
        colmax = fmaxf(colmax, fmaxf(acc[v] * gs + bv, 0.f));
    }
    colmax = fmaxf(colmax, __shfl_xor(colmax, 16, 32));
    if (cnt == 0) colmax = 0.f;
    if (lane < 16) feats[(size_t)kpg * F_DIM + 288 + ri * 64 + nt * 16 + m] = colmax;
  }
}

// ---------------------------------------------------------------- 7) fusion GEMM (WMMA fp32)
// (4096 x 416) @ (416 x 128); one wave per (16-row tile, 16-col tile).
__global__ __launch_bounds__(128) void fusion_kernel(const float* __restrict__ feats,
                                                     const float* __restrict__ W,
                                                     const float* __restrict__ gg,
                                                     const float* __restrict__ bbv,
                                                     float* __restrict__ out) {
  const int wid = threadIdx.x >> 5;
  const int lane = threadIdx.x & 31;
  const int g = blockIdx.x * 4 + wid;
  const int mt = g >> 3;
  const int nt = g & 7;
  const int m = lane & 15;
  const int koff = (lane < 16) ? 0 : 2;
  const float* Arow = feats + ((size_t)mt * 16 + m) * F_DIM;
  const int col = nt * 16 + m;
  v8f acc = zero8();
  for (int k0 = 0; k0 < F_DIM; k0 += 4) {
    const int ka = k0 + koff;
    const float2 av = *(const float2*)(Arow + ka);
    v2f a; a.x = av.x; a.y = av.y;
    v2f bm;
    bm.x = W[(size_t)(ka + 0) * OUT_DIM + col];
    bm.y = W[(size_t)(ka + 1) * OUT_DIM + col];
    acc = wmma_k4(a, bm, acc);
  }
  const float gs = gg[col] * BN_SCALE;
  const float bb = bbv[col];
  const int mbase = (lane < 16) ? 0 : 8;
#pragma unroll
  for (int v = 0; v < 8; ++v)
    out[((size_t)mt * 16 + mbase + v) * OUT_DIM + col] = fmaxf(acc[v] * gs + bb, 0.f);
}

// ---------------------------------------------------------------- launch
extern "C" void kernel_launch(void* const* d_in, const int* in_sizes, int n_in,
                              void* d_out, int out_size, void* d_ws, size_t ws_size,
                              hipStream_t stream) {
  const float* points   = (const float*)d_in[0];
  const float* spatial  = (const float*)d_in[1];
  const int*   vcoords  = (const int*)d_in[2];
  const float* vfeats   = (const float*)d_in[3];
  const float* raw_w0   = (const float*)d_in[4];
  const float* raw_g0   = (const float*)d_in[5];
  const float* raw_b0   = (const float*)d_in[6];
  const float* raw_w1   = (const float*)d_in[7];
  const float* raw_g1   = (const float*)d_in[8];
  const float* raw_b1   = (const float*)d_in[9];
  const float* conv_w0  = (const float*)d_in[10];
  const float* conv_g0  = (const float*)d_in[11];
  const float* conv_b0  = (const float*)d_in[12];
  const float* conv_w1  = (const float*)d_in[13];
  const float* conv_g1  = (const float*)d_in[14];
  const float* conv_b1  = (const float*)d_in[15];
  const float* fusion_w = (const float*)d_in[16];
  const float* fusion_g = (const float*)d_in[17];
  const float* fusion_b = (const float*)d_in[18];
  (void)in_sizes; (void)n_in; (void)out_size; (void)ws_size;

  char* ws = (char*)d_ws;
  float* kp    = (float*)(ws + 0);                       //  48 KB (B*N_KP*3)
  float* feats = (float*)(ws + 65536);                   // 4096*416 f32
  int* ridx    = (int*)(ws + 65536 + 6815744);           // 4096*32
  int* rcnt    = (int*)(ws + 65536 + 6815744 + 524288);  // 4096*2
  int* cidx    = (int*)(ws + 65536 + 6815744 + 524288 + 32768);            // 4096*48
  int* ccnt    = (int*)(ws + 65536 + 6815744 + 524288 + 32768 + 786432);   // 4096*2

  fps_kernel<<<BB, 1024, 0, stream>>>(points, kp);
  bev_kernel<<<BB * N_KP, 256, 0, stream>>>(spatial, kp, feats);
  raw_group_kernel<<<(BB * N_KP * 2) / 8, 256, 0, stream>>>(points, kp, ridx, rcnt);
  conv_group_kernel<<<(BB * N_KP * 2) / 8, 256, 0, stream>>>(vcoords, kp, cidx, ccnt);
  raw_mlp_kernel<<<(BB * N_KP * 2) / 4, 128, 0, stream>>>(
      points, kp, ridx, rcnt, raw_w0, raw_g0, raw_b0, raw_w1, raw_g1, raw_b1, feats);
  conv_mlp_kernel<<<(BB * N_KP * 2) / 2, 64, 0, stream>>>(
      vcoords, vfeats, kp, cidx, ccnt, conv_w0, conv_g0, conv_b0, conv_w1, conv_g1,
      conv_b1, feats);
  fusion_kernel<<<(256 * 8) / 4, 128, 0, stream>>>(feats, fusion_w, fusion_g, fusion_b,
                                                   (float*)d_out);
}